// Dynamic_MultiTeacher6_52699248721952
// MI455X (gfx1250) — compile-verified
//
#include <hip/hip_runtime.h>
#include <stdint.h>

#define GLOBAL_AS __attribute__((address_space(1)))
#define LDS_AS    __attribute__((address_space(3)))

typedef float v2f __attribute__((ext_vector_type(2)));
typedef float v8f __attribute__((ext_vector_type(8)));
typedef int   v4i __attribute__((vector_size(16)));   // matches builtin's pointee

static constexpr int   kB      = 4096;
static constexpr int   kC      = 1000;
static constexpr float kTkd    = 20.0f;
static constexpr float kTth    = 2.0f;
static constexpr float kAlpha  = 0.8f;
static constexpr int   kRS     = 24;   // floats per row record
static constexpr int   kRecOff = 64;   // float offset of records in ws (256 B)

// ---- ordered-uint encoding so atomicMax works on signed floats ----
__device__ __forceinline__ unsigned f2key(float f) {
  unsigned u = __float_as_uint(f);
  return (u & 0x80000000u) ? ~u : (u | 0x80000000u);
}
__device__ __forceinline__ float key2f(unsigned k) {
  unsigned u = (k & 0x80000000u) ? (k ^ 0x80000000u) : ~k;
  return __uint_as_float(u);
}

// ---- CDNA5 async global->LDS copy (ASYNCcnt path) ----
__device__ __forceinline__ void async_copy_b128(const float* g, float* l) {
#if __has_builtin(__builtin_amdgcn_global_load_async_to_lds_b128)
  __builtin_amdgcn_global_load_async_to_lds_b128(
      (GLOBAL_AS v4i*)(uintptr_t)g,
      (LDS_AS v4i*)(unsigned)(uintptr_t)l, 0, 0);
#else
  unsigned           lo = (unsigned)(uintptr_t)l;
  unsigned long long ga = (unsigned long long)(uintptr_t)g;
  asm volatile("global_load_async_to_lds_b128 %0, %1, off"
               :: "v"(lo), "v"(ga) : "memory");
#endif
}

__device__ __forceinline__ void wait_async0() {
#if __has_builtin(__builtin_amdgcn_s_wait_asynccnt)
  __builtin_amdgcn_s_wait_asynccnt(0);
#else
  asm volatile("s_wait_asynccnt 0" ::: "memory");
#endif
}

// ---- exact f32 wave(32) sum via V_WMMA_F32_16X16X4_F32 ----
// A: vgpr0 = p (lanes0-15 -> A[m,0], lanes16-31 -> A[m,2]), vgpr1 = 0.
// B: all ones => D[m,n] = p_m + p_{m+16} for every n.
// Summing a lane's 8 D vgprs: lanes 0-15 get sum over m=0..7, lanes 16-31 m=8..15.
__device__ __forceinline__ float wave_sum32(float p) {
#if __has_builtin(__builtin_amdgcn_wmma_f32_16x16x4_f32)
  v2f a; a[0] = p;    a[1] = 0.0f;
  v2f b; b[0] = 1.0f; b[1] = 1.0f;
  v8f c = {};
  v8f d = __builtin_amdgcn_wmma_f32_16x16x4_f32(false, a, false, b,
                                                (short)0, c, false, false);
  float s = d[0] + d[1] + d[2] + d[3] + d[4] + d[5] + d[6] + d[7];
  return __shfl(s, 0, 32) + __shfl(s, 16, 32);
#else
  #pragma unroll
  for (int off = 16; off; off >>= 1) p += __shfl_xor(p, off, 32);
  return p;
#endif
}

__global__ void mt6_init_kernel(unsigned* wsu) {
  if (threadIdx.x == 0) wsu[0] = 0u;  // key-space lower bound
}

// ---- Pass 1: one block per sample row; everything per-row except max_preds ----
__global__ void __launch_bounds__(256)
mt6_row_kernel(const float* __restrict__ t0, const float* __restrict__ t1,
               const float* __restrict__ t2, const float* __restrict__ t3,
               const float* __restrict__ t4, const float* __restrict__ t5,
               const float* __restrict__ outs, const int* __restrict__ targets,
               float* __restrict__ wsf, unsigned* __restrict__ wsu)
{
  __shared__ __align__(16) float sm[7][kC];   // 6 teachers + student, 28 KB
  __shared__ float redT[8][8][2];             // top2 cross-wave scratch
  __shared__ float redS[8][8][2];             // sum  cross-wave scratch

  const int  row    = blockIdx.x;
  const int  tid    = threadIdx.x;
  const int  wid    = tid >> 5;
  const int  lane   = tid & 31;
  const bool active = tid < (kC / 4);         // 250 threads own 4 columns each
  const int  c0     = tid * 4;

  // Stage all 7 rows into LDS with async b128 copies (row base is 16B aligned).
  if (active) {
    const size_t rb = (size_t)row * kC;
    async_copy_b128(t0  + rb + c0, &sm[0][c0]);
    async_copy_b128(t1  + rb + c0, &sm[1][c0]);
    async_copy_b128(t2  + rb + c0, &sm[2][c0]);
    async_copy_b128(t3  + rb + c0, &sm[3][c0]);
    async_copy_b128(t4  + rb + c0, &sm[4][c0]);
    async_copy_b128(t5  + rb + c0, &sm[5][c0]);
    async_copy_b128(outs + rb + c0, &sm[6][c0]);
  }
  wait_async0();
  __syncthreads();

  // Registers: X[0..5]=teachers, X[6]=mimic, X[7]=student. Data read once.
  float X[8][4];
  #pragma unroll
  for (int a = 0; a < 8; ++a)
    #pragma unroll
    for (int j = 0; j < 4; ++j) X[a][j] = 0.0f;
  if (active) {
    #pragma unroll
    for (int a = 0; a < 6; ++a)
      #pragma unroll
      for (int j = 0; j < 4; ++j) X[a][j] = sm[a][c0 + j];
    #pragma unroll
    for (int j = 0; j < 4; ++j) {
      X[7][j] = sm[6][c0 + j];
      X[6][j] = (X[0][j] + X[1][j] + X[2][j] + X[3][j] + X[4][j] + X[5][j])
                * (1.0f / 6.0f);
    }
  }

  // ---- top-2 (and row max) for all 8 vectors ----
  float T1[8], T2[8];
  #pragma unroll
  for (int a = 0; a < 8; ++a) {
    float p1, p2;
    if (active) {
      float m01 = fmaxf(X[a][0], X[a][1]), n01 = fminf(X[a][0], X[a][1]);
      float m23 = fmaxf(X[a][2], X[a][3]), n23 = fminf(X[a][2], X[a][3]);
      p1 = fmaxf(m01, m23);
      p2 = fmaxf(fminf(m01, m23), (m01 >= m23) ? n01 : n23);
    } else { p1 = -INFINITY; p2 = -INFINITY; }
    #pragma unroll
    for (int off = 16; off; off >>= 1) {
      float o1 = __shfl_xor(p1, off, 32);
      float o2 = __shfl_xor(p2, off, 32);
      float n1 = fmaxf(p1, o1);
      float n2 = fmaxf(fminf(p1, o1), (p1 >= o1) ? p2 : o2);
      p1 = n1; p2 = n2;
    }
    if (lane == 0) { redT[a][wid][0] = p1; redT[a][wid][1] = p2; }
  }
  __syncthreads();
  #pragma unroll
  for (int a = 0; a < 8; ++a) {
    float p1 = redT[a][0][0], p2 = redT[a][0][1];
    #pragma unroll
    for (int w = 1; w < 8; ++w) {
      float o1 = redT[a][w][0], o2 = redT[a][w][1];
      float n1 = fmaxf(p1, o1);
      float n2 = fmaxf(fminf(p1, o1), (p1 >= o1) ? p2 : o2);
      p1 = n1; p2 = n2;
    }
    T1[a] = p1; T2[a] = p2;
  }

  // ---- exp-sums: a<7: (Z_a, dot_a) at T_KD; a==7: (Z1, Z_T) for student ----
  #pragma unroll
  for (int a = 0; a < 8; ++a) {
    float z = 0.0f, d = 0.0f;
    if (active) {
      if (a < 7) {
        float m = T1[a];
        #pragma unroll
        for (int j = 0; j < 4; ++j) {
          float e = __expf((X[a][j] - m) * (1.0f / kTkd));
          z += e; d += e * X[7][j];
        }
      } else {
        float m = T1[7];
        #pragma unroll
        for (int j = 0; j < 4; ++j) {
          z += __expf(X[7][j] - m);
          d += __expf((X[7][j] - m) * (1.0f / kTkd));
        }
      }
    }
    #pragma unroll
    for (int off = 16; off; off >>= 1) {
      z += __shfl_xor(z, off, 32);
      d += __shfl_xor(d, off, 32);
    }
    if (lane == 0) { redS[a][wid][0] = z; redS[a][wid][1] = d; }
  }
  __syncthreads();

  if (tid == 0) {
    float Z[8], D[8];
    #pragma unroll
    for (int a = 0; a < 8; ++a) {
      float z = 0.0f, d = 0.0f;
      #pragma unroll
      for (int w = 0; w < 8; ++w) { z += redS[a][w][0]; d += redS[a][w][1]; }
      Z[a] = z; D[a] = d;
    }
    const int tgt = targets[row];
    float tl[7]; float msum = 0.0f;
    #pragma unroll
    for (int a = 0; a < 6; ++a) { tl[a] = sm[a][tgt]; msum += tl[a]; }
    tl[6] = msum * (1.0f / 6.0f);
    const float s_tgt = sm[6][tgt];
    const float m_s   = T1[7];
    const float CE    = (m_s + __logf(Z[7])) - s_tgt;                 // -logp[tgt]
    const float lseT  = m_s * (1.0f / kTkd) + __logf(D[7]);           // LSE of s/T

    float* rec = wsf + kRecOff + (size_t)row * kRS;
    #pragma unroll
    for (int a = 0; a < 7; ++a) {
      rec[a]      = (tl[a] == T1[a]) ? (T1[a] - T2[a]) : 0.0f;        // margin
      rec[7 + a]  = kTkd * kTkd * lseT - kTkd * (D[a] / Z[a]);        // KD_t
      rec[14 + a] = tl[a];                                            // tgt logit
    }
    rec[21] = CE;
    float bm = fmaxf(fmaxf(fmaxf(T1[0], T1[1]), fmaxf(T1[2], T1[3])),
                     fmaxf(T1[4], T1[5]));
    atomicMax(wsu, f2key(bm));   // order-independent -> deterministic
  }
}

// ---- Pass 2: weight + reduce 4096 row records to the scalar loss ----
__global__ void __launch_bounds__(256)
mt6_final_kernel(const float* __restrict__ wsf, const unsigned* __restrict__ wsu,
                 float* __restrict__ out)
{
  __shared__ float wsum[8];
  const float inv_maxp = 1.0f / key2f(wsu[0]);
  float acc = 0.0f;
  for (int r = threadIdx.x; r < kB; r += 256) {
    const float* rec = wsf + kRecOff + (size_t)r * kRS;
    float m[7], kd[7], tg[7];
    #pragma unroll
    for (int t = 0; t < 7; ++t) { m[t] = rec[t]; kd[t] = rec[7 + t]; tg[t] = rec[14 + t]; }
    const float CE = rec[21];
    float mx = m[0];
    #pragma unroll
    for (int t = 1; t < 7; ++t) mx = fmaxf(mx, m[t]);
    float e[7]; float zs = 0.0f;
    #pragma unroll
    for (int t = 0; t < 7; ++t) { e[t] = __expf((m[t] - mx) * (1.0f / kTth)); zs += e[t]; }
    const float invz = 1.0f / zs;
    float l = 0.0f;
    #pragma unroll
    for (int t = 0; t < 7; ++t) {
      float w2 = kAlpha * tg[t] * inv_maxp;
      float w1 = 1.0f - w2;
      l += (e[t] * invz) * (w1 * CE + w2 * kd[t]);
    }
    acc += l;
  }
  float s = wave_sum32(acc);     // v_wmma_f32_16x16x4_f32 reduction
  const int wid = threadIdx.x >> 5, lane = threadIdx.x & 31;
  if (lane == 0) wsum[wid] = s;
  __syncthreads();
  if (threadIdx.x == 0) {
    float t = 0.0f;
    #pragma unroll
    for (int w = 0; w < 8; ++w) t += wsum[w];
    out[0] = t * (1.0f / (float)kB);
  }
}

extern "C" void kernel_launch(void* const* d_in, const int* in_sizes, int n_in,
                              void* d_out, int out_size, void* d_ws, size_t ws_size,
                              hipStream_t stream) {
  const float* t0   = (const float*)d_in[0];
  const float* t1   = (const float*)d_in[1];
  const float* t2   = (const float*)d_in[2];
  const float* t3   = (const float*)d_in[3];
  const float* t4   = (const float*)d_in[4];
  const float* t5   = (const float*)d_in[5];
  const float* outs = (const float*)d_in[6];
  const int*   tg   = (const int*)d_in[7];
  float*    wsf = (float*)d_ws;
  unsigned* wsu = (unsigned*)d_ws;

  mt6_init_kernel<<<1, 64, 0, stream>>>(wsu);
  mt6_row_kernel<<<kB, 256, 0, stream>>>(t0, t1, t2, t3, t4, t5, outs, tg, wsf, wsu);
  mt6_final_kernel<<<1, 256, 0, stream>>>(wsf, wsu, (float*)d_out);
}